// QuantumFullyConnectedLayer_65481071405380
// MI455X (gfx1250) — compile-verified
//
#include <hip/hip_runtime.h>

// CDNA5 / gfx1250: wave32, WMMA 16x16x4 f32.
typedef float v2f __attribute__((ext_vector_type(2)));
typedef float v8f __attribute__((ext_vector_type(8)));

// out[t] = bias + sum_q cos(x[t*16+q]) * (cos(theta[q]) * w[q])
//
// WMMA mapping (D = A(16x4) x B(4x16) + C), one chunk k covers q = 4k..4k+3:
//   A[m][kk] = c[4k+kk]  (broadcast over rows m)  -> every row of D identical
//   B[kk][n] = cos(x[token_n, 4k+kk])             -> tokens along N
//   C seeded with bias; 4 chained WMMAs accumulate q = 0..15.
// VGPR fragments (wave32):
//   A: vgpr0 = K{0|2} (lane<16 | lane>=16), vgpr1 = K{1|3}  -> a[k] = {c[4k+h], c[4k+h+1]}
//   B: vgpr0 = row K{0|2}, col N = lane%16; vgpr1 = K{1|3}  -> matches float2 loads below
//   D: vgpr0 lane l -> row 8*(l/16), col l%16  => d0[lane] = out[token (lane%16)]
__global__ __launch_bounds__(256) void qfc16_wmma_kernel(
    const float* __restrict__ x,
    const float* __restrict__ theta,
    const float* __restrict__ w,
    const float* __restrict__ bias,
    float* __restrict__ out,
    int ntokens)
{
    const int lane = threadIdx.x & 31;
    const int n    = lane & 15;           // column within the 16-token group
    const int h    = (lane >> 4) << 1;    // 0 for lanes 0-15, 2 for lanes 16-31

    // Per-lane A fragments: combined weights c[q] = cos(theta[q]) * w[q]
    v2f a[4];
#pragma unroll
    for (int k = 0; k < 4; ++k) {
        const int q = 4 * k + h;
        a[k][0] = __cosf(theta[q])     * w[q];
        a[k][1] = __cosf(theta[q + 1]) * w[q + 1];
    }

    const float bv = bias[0];
    v8f cbias;
#pragma unroll
    for (int i = 0; i < 8; ++i) cbias[i] = bv;

    const int wave   = (int)((blockIdx.x * blockDim.x + threadIdx.x) >> 5);
    const int nwaves = (int)((gridDim.x * blockDim.x) >> 5);
    const int ntiles = ntokens >> 5;      // 32 tokens per wave-iteration

    for (int tile = wave; tile < ntiles; tile += nwaves) {
        const float* base = x + (size_t)tile * (32 * 16);
        // set 0: tokens [tile*32 .. +15], set 1: tokens [tile*32+16 .. +31]
        const float* p0 = base + n * 16 + h;
        const float* p1 = p0 + 16 * 16;

        v2f r0[4], r1[4];
#pragma unroll
        for (int k = 0; k < 4; ++k) {
            r0[k] = *(const v2f*)(p0 + 4 * k);   // global_load_b64, q = {4k+h, 4k+h+1}
            r1[k] = *(const v2f*)(p1 + 4 * k);
        }

        v8f acc0 = cbias, acc1 = cbias;
#pragma unroll
        for (int k = 0; k < 4; ++k) {
            v2f b0, b1;
            b0[0] = __cosf(r0[k][0]);  b0[1] = __cosf(r0[k][1]);
            b1[0] = __cosf(r1[k][0]);  b1[1] = __cosf(r1[k][1]);
            acc0 = __builtin_amdgcn_wmma_f32_16x16x4_f32(
                false, a[k], false, b0, (short)0, acc0, false, false);
            acc1 = __builtin_amdgcn_wmma_f32_16x16x4_f32(
                false, a[k], false, b1, (short)0, acc1, false, false);
        }

        // D vgpr0: lanes 0-15 hold out[token n] (set0), lanes 16-31 hold
        // out[token n] of set1 -> lane l stores out[tile*32 + l]; 128B coalesced.
        const float res = (lane < 16) ? acc0[0] : acc1[0];
        out[(size_t)tile * 32 + lane] = res;
    }
}

extern "C" void kernel_launch(void* const* d_in, const int* in_sizes, int n_in,
                              void* d_out, int out_size, void* d_ws, size_t ws_size,
                              hipStream_t stream)
{
    const float* x     = (const float*)d_in[0];   // [128, 8192, 16] f32
    const float* theta = (const float*)d_in[1];   // [16]
    const float* w     = (const float*)d_in[2];   // [1, 16]
    const float* b     = (const float*)d_in[3];   // [1]
    float* out = (float*)d_out;                   // [128, 8192, 1] f32

    const int ntokens = out_size;                 // 1,048,576 tokens
    const int threads = 256;                      // 8 wave32s per block
    const int wavesPerBlock = threads / 32;
    const int ntiles = ntokens / 32;
    int blocks = (ntiles + wavesPerBlock - 1) / wavesPerBlock;
    if (blocks > 2048) blocks = 2048;             // grid-stride; ~16K waves saturates HBM

    qfc16_wmma_kernel<<<blocks, threads, 0, stream>>>(x, theta, w, b, out, ntokens);
}